// TripletLoss_14800457302034
// MI455X (gfx1250) — compile-verified
//
#include <hip/hip_runtime.h>
#include <hip/hip_bf16.h>

typedef float v2f __attribute__((ext_vector_type(2)));
typedef float v8f __attribute__((ext_vector_type(8)));

#define N_ROWS 512
#define DIM    16384
#define NTRIP  1536
#define NPAIR  2560           // 512 norms + 512 (i,j) + 1536 (i,k)
#define NTILE  (NPAIR / 16)   // 160 tiles of 16 pairs
#define KSEG   (DIM / 8)      // K-span per wave (8 waves per block)

// Build gather lists of (a,b) row pairs whose dot products we need.
__global__ void build_pairs_kernel(const int* __restrict__ trip,
                                   int* __restrict__ idxA,
                                   int* __restrict__ idxB) {
    int p = blockIdx.x * blockDim.x + threadIdx.x;
    if (p >= NPAIR) return;
    int a, b;
    if (p < 512) {                       // norms: x_p . x_p
        a = p; b = p;
    } else if (p < 1024) {               // x_i . x_j (j shared across i's 3 triplets)
        int t = (p - 512) * 3;           // first triplet of row i
        a = trip[t * 3 + 0];
        b = trip[t * 3 + 1];
    } else {                             // x_i . x_k per triplet
        int t = p - 1024;
        a = trip[t * 3 + 0];
        b = trip[t * 3 + 2];
    }
    idxA[p] = a;
    idxB[p] = b;
}

// One block per 16-pair tile. 8 waves split the K dimension; each wave runs a
// chain of V_WMMA_F32_16X16X4_F32, then partial C tiles are reduced in LDS and
// the diagonal is extracted (dot products of the gathered pairs).
__global__ __launch_bounds__(256) void wmma_dots_kernel(
        const float* __restrict__ X,
        const int* __restrict__ idxA,
        const int* __restrict__ idxB,
        float* __restrict__ dots) {
    __shared__ float red[8 * 256];

    const int tile = blockIdx.x;
    const int tid  = threadIdx.x;
    const int wave = tid >> 5;
    const int lane = tid & 31;

    // fp32 16x16x4 A fragment layout: lanes 0-15 hold K=0,1 ; lanes 16-31 hold K=2,3.
    const int m     = lane & 15;          // A row / B column within tile
    const int khalf = (lane >> 4) * 2;    // 0 or 2

    const int rowA = idxA[tile * 16 + m];
    const int rowB = idxB[tile * 16 + m];
    const float* pa = X + (long)rowA * DIM + khalf;
    const float* pb = X + (long)rowB * DIM + khalf;

    const int k0 = wave * KSEG;
    const int k1 = k0 + KSEG;

    v8f c = {0.f, 0.f, 0.f, 0.f, 0.f, 0.f, 0.f, 0.f};
    #pragma unroll 4
    for (int k = k0; k < k1; k += 4) {
        v2f a = *(const v2f*)(pa + k);    // 8B-aligned global_load_b64
        v2f b = *(const v2f*)(pb + k);
        c = __builtin_amdgcn_wmma_f32_16x16x4_f32(
                /*neg_a=*/false, a, /*neg_b=*/false, b,
                /*c_mod=*/(short)0, c, /*reuse_a=*/false, /*reuse_b=*/false);
    }

    // Stage this wave's C tile in LDS.
    #pragma unroll
    for (int v = 0; v < 8; ++v)
        red[wave * 256 + lane * 8 + v] = c[v];
    __syncthreads();

    // Reduce the 8 partial tiles: each thread owns one (lane, vgpr) slot.
    float s = 0.f;
    #pragma unroll
    for (int w = 0; w < 8; ++w)
        s += red[w * 256 + tid];

    // Diagonal slots in the (lane*8 + v) flattening of the C/D layout:
    //   m in [0,8):  lane=m,    v=m   -> slot 9m
    //   m in [8,16): lane=m+16, v=m-8 -> slot 9m + 120
    if (tid < 72 && tid % 9 == 0)
        dots[tile * 16 + tid / 9] = s;
    if (tid >= 192 && (tid - 120) % 9 == 0)
        dots[tile * 16 + (tid - 120) / 9] = s;
}

// Single block: assemble distances, softplus, mean.
__global__ __launch_bounds__(256) void finalize_kernel(
        const int* __restrict__ trip,
        const float* __restrict__ dots,
        float* __restrict__ out) {
    __shared__ float red[256];
    const int tid = threadIdx.x;

    float acc = 0.f;
    for (int t = tid; t < NTRIP; t += 256) {
        int i = trip[t * 3 + 0];
        int j = trip[t * 3 + 1];
        int k = trip[t * 3 + 2];
        float ni = dots[i], nj = dots[j], nk = dots[k];
        float dij = fmaxf(ni + nj - 2.f * dots[512 + i], 0.f);
        float dik = fmaxf(ni + nk - 2.f * dots[1024 + t], 0.f);
        float x = dij - dik;
        // numerically stable softplus
        acc += fmaxf(x, 0.f) + log1pf(expf(-fabsf(x)));
    }
    red[tid] = acc;
    __syncthreads();
    #pragma unroll
    for (int s = 128; s > 0; s >>= 1) {
        if (tid < s) red[tid] += red[tid + s];
        __syncthreads();
    }
    if (tid == 0) out[0] = red[0] / (float)NTRIP;
}

extern "C" void kernel_launch(void* const* d_in, const int* in_sizes, int n_in,
                              void* d_out, int out_size, void* d_ws, size_t ws_size,
                              hipStream_t stream) {
    const float* X    = (const float*)d_in[0];   // features (512, 16384) fp32
    const int*   trip = (const int*)d_in[1];     // triplets (1536, 3) int32
    float* out = (float*)d_out;

    // Workspace layout: dots[2560] | idxA[2560] | idxB[2560]  (~30 KB)
    float* dots = (float*)d_ws;
    int*   idxA = (int*)(dots + NPAIR);
    int*   idxB = idxA + NPAIR;

    build_pairs_kernel<<<(NPAIR + 255) / 256, 256, 0, stream>>>(trip, idxA, idxB);
    wmma_dots_kernel<<<NTILE, 256, 0, stream>>>(X, idxA, idxB, dots);
    finalize_kernel<<<1, 256, 0, stream>>>(trip, dots, out);
}